// StreamingPCENTransform_76613626626312
// MI455X (gfx1250) — compile-verified
//
#include <hip/hip_runtime.h>
#include <hip/hip_bf16.h>

typedef __attribute__((ext_vector_type(2))) float v2f;
typedef __attribute__((ext_vector_type(8))) float v8f;

#define PCEN_EPS 1e-6f
#define T_DIM 2048
#define F_DIM 257
#define BM_DIM 64              // B*MICS = 16*4
#define FTILES 17              // ceil(257/16)
#define NTASKS (BM_DIM * FTILES)
#define WAVES_PER_BLOCK 8

// Blocked-scan PCEN: per wave, one (bm, 16-freq) chain. Each 16-time tile is
//   M_tile = L * X_tile + carry,   L[j][i] = (i<=j) ? s*a^(j-i) : 0,
//   carry[j][f] = a^(j+1) * m_prev[f]
// computed with 5 chained V_WMMA_F32_16X16X4_F32 ops (4 K-slices of L + rank-1 carry).
__global__ __launch_bounds__(256) void StreamingPCEN_wmma_kernel(
    const float* __restrict__ x,
    const float* __restrict__ s_p,
    const float* __restrict__ alpha_p,
    const float* __restrict__ delta_p,
    const float* __restrict__ r_p,
    float* __restrict__ out)
{
    const int lane = threadIdx.x & 31;
    const int wave = threadIdx.x >> 5;
    const int task = blockIdx.x * WAVES_PER_BLOCK + wave;
    if (task >= NTASKS) return;          // wave-uniform exit (EXEC stays all-1s for WMMA)

    const int bmi = task / FTILES;
    const int ft  = task % FTILES;
    const int f0  = ft * 16;

    const int n    = lane & 15;          // column (freq) / row-in-half index
    const int half = lane >> 4;          // 0: K0/K1, M0-7 ; 1: K2/K3, M8-15

    const int  f      = f0 + n;
    const int  fc     = (f < F_DIM) ? f : (F_DIM - 1);  // clamp loads for ragged last tile
    const bool fvalid = (f < F_DIM);

    const float s0    = s_p[0];
    const float a     = 1.0f - s0;
    const float alpha = alpha_p[0];
    const float delta = delta_p[0];
    const float rr    = r_p[0];
    const float dr    = __expf(rr * __logf(delta));     // delta^r

    // A-operand (16x4 f32): lanes 0-15: M=n, v.x=K(4k), v.y=K(4k+1);
    //                       lanes 16-31: M=n, v.x=K(4k+2), v.y=K(4k+3)
    v2f L[4];
#pragma unroll
    for (int k = 0; k < 4; ++k) {
        const int i0 = 4 * k + 2 * half;
        const int i1 = i0 + 1;
        L[k].x = (i0 <= n) ? s0 * __powf(a, (float)(n - i0)) : 0.0f;
        L[k].y = (i1 <= n) ? s0 * __powf(a, (float)(n - i1)) : 0.0f;
    }
    // Carry A-operand: A[j][0] = a^(j+1), other columns zero.
    v2f AW;
    AW.x = (half == 0) ? __powf(a, (float)(n + 1)) : 0.0f;
    AW.y = 0.0f;

    const long   base = (long)bmi * T_DIM * F_DIM;
    const float* xrow = x + base;
    float*       orow = out + base;

    // Virtual m_{-1} = x_0  ->  m_0 = a*x0 + s*x0 = x0 (reference first-frame branch).
    float mprev = xrow[fc];

    for (int t0 = 0; t0 < T_DIM; t0 += 16) {
        // B-layout X loads (K x N): lane n, half h holds K = 4k+2h, 4k+2h+1 of column f.
        v2f Bk[4];
#pragma unroll
        for (int k = 0; k < 4; ++k) {
            const int tb = t0 + 4 * k + 2 * half;
            Bk[k].x = xrow[(long)tb * F_DIM + fc];
            Bk[k].y = xrow[(long)(tb + 1) * F_DIM + fc];
        }
        // D-layout X reload for the pointwise stage (hits L0/L2, HBM traffic stays ~1x).
        float xd[8];
#pragma unroll
        for (int rI = 0; rI < 8; ++rI) {
            xd[rI] = xrow[(long)(t0 + rI + 8 * half) * F_DIM + fc];
        }

        // Carry B-operand: row 0 = m_prev[f], rows 1-3 zero.
        v2f Bc;
        Bc.x = (half == 0) ? mprev : 0.0f;
        Bc.y = 0.0f;

        v8f d = {0.f, 0.f, 0.f, 0.f, 0.f, 0.f, 0.f, 0.f};
        d = __builtin_amdgcn_wmma_f32_16x16x4_f32(false, AW, false, Bc, (short)0, d, false, false);
#pragma unroll
        for (int k = 0; k < 4; ++k) {
            d = __builtin_amdgcn_wmma_f32_16x16x4_f32(false, L[k], false, Bk[k], (short)0, d, false, false);
        }

        // Pointwise PCEN: (x/(m+eps)^alpha + delta)^r - delta^r  (inputs > 0, logs safe).
#pragma unroll
        for (int rI = 0; rI < 8; ++rI) {
            const float m  = d[rI];
            const float dn = __expf(alpha * __logf(m + PCEN_EPS));
            const float v  = __expf(rr * __logf(xd[rI] / dn + delta)) - dr;
            if (fvalid) {
                orow[(long)(t0 + rI + 8 * half) * F_DIM + f] = v;
            }
        }

        // Carry m_{t0+15}: lives in upper-half lanes, d[7]. Broadcast upper -> lower half.
        mprev = __shfl(d[7], n + 16, 32);

        // Prefetch next tile's rows (global_prefetch_b8): lane -> row t0+16+(lane&15).
        if (t0 + 32 <= T_DIM) {
            __builtin_prefetch(&xrow[(long)(t0 + 16 + (lane & 15)) * F_DIM], 0, 1);
        }
    }
}

extern "C" void kernel_launch(void* const* d_in, const int* in_sizes, int n_in,
                              void* d_out, int out_size, void* d_ws, size_t ws_size,
                              hipStream_t stream) {
    const float* x     = (const float*)d_in[0];
    const float* s     = (const float*)d_in[1];
    const float* alpha = (const float*)d_in[2];
    const float* delta = (const float*)d_in[3];
    const float* r     = (const float*)d_in[4];
    float*       out   = (float*)d_out;

    dim3 grid((NTASKS + WAVES_PER_BLOCK - 1) / WAVES_PER_BLOCK);  // 136 blocks
    dim3 block(32 * WAVES_PER_BLOCK);                              // 8 waves/block
    hipLaunchKernelGGL(StreamingPCEN_wmma_kernel, grid, block, 0, stream,
                       x, s, alpha, delta, r, out);
}